// GraphSAGE_GCN_38714835206722
// MI455X (gfx1250) — compile-verified
//
#include <hip/hip_runtime.h>

#define FDIM 64
#define TILE_M 16
#define LDA 68            // padded LDS row stride (floats) -> conflict-free float2 reads
#define GRAPHS 128
#define LN_EPS 1e-5f

typedef float v2f __attribute__((ext_vector_type(2)));
typedef float v8f __attribute__((ext_vector_type(8)));

// ---------------- degree / counts ----------------
__global__ void degree_kernel(const int* __restrict__ dst, float* __restrict__ deg, int nedges) {
    int e = blockIdx.x * blockDim.x + threadIdx.x;
    if (e < nedges) atomicAdd(&deg[dst[e]], 1.0f);
}

__global__ void count_kernel(const int* __restrict__ batch, float* __restrict__ cnt, int nrows) {
    int n = blockIdx.x * blockDim.x + threadIdx.x;
    if (n < nrows) atomicAdd(&cnt[batch[n]], 1.0f);
}

__global__ void invdeg_kernel(float* __restrict__ deg, int nrows) {
    int n = blockIdx.x * blockDim.x + threadIdx.x;
    if (n < nrows) deg[n] = 1.0f / fmaxf(deg[n], 1.0f);
}

// ---------------- edge scatter (mean numerator) ----------------
// one thread per (edge, feature): gather row of src, atomic-add into agg[dst]
__global__ void scatter_add_kernel(const float* __restrict__ xin,
                                   const int* __restrict__ src, const int* __restrict__ dst,
                                   float* __restrict__ agg, int nedges) {
    int t = blockIdx.x * blockDim.x + threadIdx.x;
    int e = t >> 6;
    int f = t & 63;
    if (e < nedges) {
        int s = src[e], d = dst[e];
        atomicAdd(&agg[d * FDIM + f], xin[s * FDIM + f]);
    }
}

// ---------------- fused WMMA GEMM ----------------
// out[n,:] = (A1[n,:]*rowscale[n]) @ W1 + (A2 ? A2[n,:] @ W2 : 0) + bias + add1 + add2
// block: 128 threads (4 waves), 16 output rows; wave w computes columns [16w,16w+16)
__global__ __launch_bounds__(128)
void fused_gemm_kernel(const float* __restrict__ A1, const float* __restrict__ rowscale,
                       const float* __restrict__ W1,
                       const float* __restrict__ A2, const float* __restrict__ W2,
                       const float* __restrict__ bias,
                       const float* __restrict__ add1, const float* __restrict__ add2,
                       float* __restrict__ out, int nrows) {
    __shared__ float sA1[TILE_M * LDA];
    __shared__ float sA2[TILE_M * LDA];
    __shared__ float sW1[FDIM * LDA];   // transposed: sW1[c*LDA + k] = W1[k*FDIM + c]
    __shared__ float sW2[FDIM * LDA];

    const int tid  = threadIdx.x;
    const int wave = tid >> 5;
    const int lane = tid & 31;
    const int m0   = blockIdx.x * TILE_M;

    // stage weights transposed (coalesced global reads)
    for (int i = tid; i < FDIM * FDIM; i += 128) {
        int k = i >> 6, c = i & 63;
        sW1[c * LDA + k] = W1[i];
        if (W2) sW2[c * LDA + k] = W2[i];
    }
    // stage A tiles (16 x 64), apply per-row scale to A1 while staging
    for (int i = tid; i < TILE_M * FDIM; i += 128) {
        int r = i >> 6, c = i & 63;
        int row = m0 + r;
        float a1 = 0.0f, a2 = 0.0f;
        if (row < nrows) {
            a1 = A1[row * FDIM + c];
            if (rowscale) a1 *= rowscale[row];
            if (A2) a2 = A2[row * FDIM + c];
        }
        sA1[r * LDA + c] = a1;
        if (A2) sA2[r * LDA + c] = a2;
    }
    __syncthreads();

    v8f acc = {};
    const int c0   = wave * 16;
    const int half = lane >> 4;     // 0: K0/K1 rows 0-7 of D ; 1: K2/K3 rows 8-15
    const int l16  = lane & 15;

    // chain 1: A1 @ W1
    for (int k = 0; k < FDIM; k += 4) {
        int kk = k + 2 * half;
        v2f a, b;
        a.x = sA1[l16 * LDA + kk];
        a.y = sA1[l16 * LDA + kk + 1];
        b.x = sW1[(c0 + l16) * LDA + kk];
        b.y = sW1[(c0 + l16) * LDA + kk + 1];
        acc = __builtin_amdgcn_wmma_f32_16x16x4_f32(false, a, false, b, (short)0, acc, false, false);
    }
    // chain 2: A2 @ W2 (uniform branch, EXEC stays all-ones)
    if (A2) {
        for (int k = 0; k < FDIM; k += 4) {
            int kk = k + 2 * half;
            v2f a, b;
            a.x = sA2[l16 * LDA + kk];
            a.y = sA2[l16 * LDA + kk + 1];
            b.x = sW2[(c0 + l16) * LDA + kk];
            b.y = sW2[(c0 + l16) * LDA + kk + 1];
            acc = __builtin_amdgcn_wmma_f32_16x16x4_f32(false, a, false, b, (short)0, acc, false, false);
        }
    }

    // epilogue: D layout -> lane half selects rows 0-7 / 8-15, l16 selects column
    const int c = c0 + l16;
    const float bv = bias ? bias[c] : 0.0f;
    for (int r = 0; r < 8; ++r) {
        int row = m0 + half * 8 + r;
        if (row < nrows) {
            int idx = row * FDIM + c;
            float v = acc[r] + bv;
            if (add1) v += add1[idx];
            if (add2) v += add2[idx];
            out[idx] = v;
        }
    }
}

// ---------------- graph-LN stats: wave per node ----------------
__global__ __launch_bounds__(256)
void ln_stats_kernel(const float* __restrict__ pre, const int* __restrict__ batch,
                     float* __restrict__ gsum, float* __restrict__ gsumsq, int nrows) {
    int node = blockIdx.x * (blockDim.x >> 5) + (threadIdx.x >> 5);
    int lane = threadIdx.x & 31;
    if (node >= nrows) return;
    const float2* row = (const float2*)(pre + (size_t)node * FDIM);
    float2 v = row[lane];
    float s  = v.x + v.y;
    float ss = v.x * v.x + v.y * v.y;
    for (int off = 16; off > 0; off >>= 1) {
        s  += __shfl_down(s, off, 32);
        ss += __shfl_down(ss, off, 32);
    }
    if (lane == 0) {
        int g = batch[node];
        atomicAdd(&gsum[g], s);
        atomicAdd(&gsumsq[g], ss);
    }
}

__global__ void ln_finalize_kernel(const float* __restrict__ gsum, const float* __restrict__ gsumsq,
                                   const float* __restrict__ cnt,
                                   float* __restrict__ gmean, float* __restrict__ grstd, int ngraphs) {
    int g = blockIdx.x * blockDim.x + threadIdx.x;
    if (g < ngraphs) {
        float norm = fmaxf(cnt[g], 1.0f) * (float)FDIM;
        float m   = gsum[g] / norm;
        float var = fmaxf(gsumsq[g] / norm - m * m, 0.0f);
        gmean[g] = m;
        grstd[g] = rsqrtf(var + LN_EPS);
    }
}

__global__ void ln_prelu_kernel(const float* __restrict__ pre, const int* __restrict__ batch,
                                const float* __restrict__ gmean, const float* __restrict__ grstd,
                                const float* __restrict__ w, const float* __restrict__ b,
                                const float* __restrict__ alpha,
                                float* __restrict__ out, int nrows) {
    int t = blockIdx.x * blockDim.x + threadIdx.x;
    int n = t >> 6;
    int f = t & 63;
    if (n < nrows) {
        int g = batch[n];
        float v = (pre[n * FDIM + f] - gmean[g]) * grstd[g] * w[f] + b[f];
        out[n * FDIM + f] = (v >= 0.0f) ? v : alpha[0] * v;
    }
}

// ---------------- host orchestration ----------------
extern "C" void kernel_launch(void* const* d_in, const int* in_sizes, int n_in,
                              void* d_out, int out_size, void* d_ws, size_t ws_size,
                              hipStream_t stream) {
    (void)n_in; (void)out_size; (void)ws_size;

    const float* x    = (const float*)d_in[0];
    const float* c0Wl = (const float*)d_in[1];
    const float* c0bl = (const float*)d_in[2];
    const float* c0Wr = (const float*)d_in[3];
    const float* c1Wl = (const float*)d_in[4];
    const float* c1bl = (const float*)d_in[5];
    const float* c1Wr = (const float*)d_in[6];
    const float* c2Wl = (const float*)d_in[7];
    const float* c2bl = (const float*)d_in[8];
    const float* c2Wr = (const float*)d_in[9];
    const float* s0W  = (const float*)d_in[10];
    const float* s1W  = (const float*)d_in[11];
    const float* ln0w = (const float*)d_in[12];
    const float* ln0b = (const float*)d_in[13];
    const float* ln1w = (const float*)d_in[14];
    const float* ln1b = (const float*)d_in[15];
    const float* ln2w = (const float*)d_in[16];
    const float* ln2b = (const float*)d_in[17];
    const float* p0a  = (const float*)d_in[18];
    const float* p1a  = (const float*)d_in[19];
    const float* p2a  = (const float*)d_in[20];
    const int*   ei   = (const int*)d_in[21];
    const int*   batch= (const int*)d_in[22];

    const int N = in_sizes[0] / FDIM;
    const int E = in_sizes[21] / 2;
    const int* src = ei;
    const int* dst = ei + E;

    // workspace carve-out
    char* ws = (char*)d_ws;
    size_t off = 0;
    auto carve = [&](size_t bytes) -> void* {
        void* p = ws + off;
        off = (off + bytes + 255) & ~(size_t)255;
        return p;
    };
    float* agg    = (float*)carve((size_t)N * FDIM * sizeof(float));
    float* pre    = (float*)carve((size_t)N * FDIM * sizeof(float));
    float* h1     = (float*)carve((size_t)N * FDIM * sizeof(float));
    float* h2     = (float*)carve((size_t)N * FDIM * sizeof(float));
    float* tmp    = (float*)carve((size_t)N * FDIM * sizeof(float));
    float* invdeg = (float*)carve((size_t)N * sizeof(float));
    float* cnt    = (float*)carve(GRAPHS * sizeof(float));
    float* gsum   = (float*)carve(2 * GRAPHS * sizeof(float));   // gsum | gsumsq
    float* gsumsq = gsum + GRAPHS;
    float* gmean  = (float*)carve(2 * GRAPHS * sizeof(float));   // gmean | grstd
    float* grstd  = gmean + GRAPHS;

    const int nb_e   = (E + 255) / 256;
    const int nb_n   = (N + 255) / 256;
    const int nb_nf  = (N * FDIM + 255) / 256;
    const int nb_ef  = (int)(((long long)E * FDIM + 255) / 256);
    const int nb_gem = (N + TILE_M - 1) / TILE_M;
    const int nb_st  = (N + 7) / 8;

    // degrees & graph counts (once)
    hipMemsetAsync(invdeg, 0, (size_t)N * sizeof(float), stream);
    hipMemsetAsync(cnt, 0, GRAPHS * sizeof(float), stream);
    degree_kernel<<<nb_e, 256, 0, stream>>>(dst, invdeg, E);
    count_kernel<<<nb_n, 256, 0, stream>>>(batch, cnt, N);
    invdeg_kernel<<<nb_n, 256, 0, stream>>>(invdeg, N);

    auto run_ln = [&](const float* pin, const float* lw, const float* lb,
                      const float* pa, float* pout) {
        hipMemsetAsync(gsum, 0, 2 * GRAPHS * sizeof(float), stream);
        ln_stats_kernel<<<nb_st, 256, 0, stream>>>(pin, batch, gsum, gsumsq, N);
        ln_finalize_kernel<<<1, GRAPHS, 0, stream>>>(gsum, gsumsq, cnt, gmean, grstd, GRAPHS);
        ln_prelu_kernel<<<nb_nf, 256, 0, stream>>>(pin, batch, gmean, grstd, lw, lb, pa, pout, N);
    };

    // ---- layer 0: h1 = PReLU(LN(SAGE(x))) ----
    hipMemsetAsync(agg, 0, (size_t)N * FDIM * sizeof(float), stream);
    scatter_add_kernel<<<nb_ef, 256, 0, stream>>>(x, src, dst, agg, E);
    fused_gemm_kernel<<<nb_gem, 128, 0, stream>>>(agg, invdeg, c0Wl, x, c0Wr, c0bl,
                                                  nullptr, nullptr, pre, N);
    run_ln(pre, ln0w, ln0b, p0a, h1);

    // ---- in2 = h1 + x @ s0_W ----
    fused_gemm_kernel<<<nb_gem, 128, 0, stream>>>(x, nullptr, s0W, nullptr, nullptr, nullptr,
                                                  h1, nullptr, tmp, N);

    // ---- layer 1: h2 = PReLU(LN(SAGE(in2))) ----
    hipMemsetAsync(agg, 0, (size_t)N * FDIM * sizeof(float), stream);
    scatter_add_kernel<<<nb_ef, 256, 0, stream>>>(tmp, src, dst, agg, E);
    fused_gemm_kernel<<<nb_gem, 128, 0, stream>>>(agg, invdeg, c1Wl, tmp, c1Wr, c1bl,
                                                  nullptr, nullptr, pre, N);
    run_ln(pre, ln1w, ln1b, p1a, h2);

    // ---- in3 = h1 + h2 + x @ s1_W ----
    fused_gemm_kernel<<<nb_gem, 128, 0, stream>>>(x, nullptr, s1W, nullptr, nullptr, nullptr,
                                                  h1, h2, tmp, N);

    // ---- layer 2: out = PReLU(LN(SAGE(in3))) ----
    hipMemsetAsync(agg, 0, (size_t)N * FDIM * sizeof(float), stream);
    scatter_add_kernel<<<nb_ef, 256, 0, stream>>>(tmp, src, dst, agg, E);
    fused_gemm_kernel<<<nb_gem, 128, 0, stream>>>(agg, invdeg, c2Wl, tmp, c2Wr, c2bl,
                                                  nullptr, nullptr, pre, N);
    run_ln(pre, ln2w, ln2b, p2a, (float*)d_out);
}